// PointNet2_38439957299474
// MI455X (gfx1250) — compile-verified
//
#include <hip/hip_runtime.h>
#include <hip/hip_bf16.h>

typedef float v2f __attribute__((ext_vector_type(2)));
typedef float v8f __attribute__((ext_vector_type(8)));

#define EPS_BN 1e-5f

// ---------------------------------------------------------------------------
// GEMM with f32 WMMA: C[rows,Cout] = A[rows,Cin] * W[Cout,Cin]^T + bias
// One wave32 computes a 16(M)x32(N) tile via V_WMMA_F32_16X16X4_F32:
// two N-tiles share the A fragment (halves A traffic, 2 independent WMMAs).
// rows is a multiple of 16, Cout a multiple of 32; K remainder (Cin%4) is
// handled in a single guarded tail step outside the hot loop so the main
// loop has NO divergent/predicated loads.
// A layout (16x4 f32): lane L -> M=L%16, VGPR r -> K = r + 2*(L/16)
// B layout (4x16 f32): lane L -> N=L%16, VGPR r -> K = r + 2*(L/16)
// C layout (16x16 f32): VGPR r, lane L -> M = r + 8*(L/16), N = L%16
// ---------------------------------------------------------------------------
__global__ __launch_bounds__(256) void gemm_bias_wmma(
    const float* __restrict__ A, const float* __restrict__ W,
    const float* __restrict__ bias, float* __restrict__ C,
    int rows, int Cin, int Cout) {
  const int wave = (int)((blockIdx.x * blockDim.x + threadIdx.x) >> 5);
  const int lane = threadIdx.x & 31;
  const int tilesM = rows >> 4;
  const int tilesN = Cout >> 5;  // 32-wide N tiles
  if (wave >= tilesM * tilesN) return;
  const int tm = wave % tilesM;
  const int tn = wave / tilesM;
  const int m0 = tm << 4, n0 = tn << 5;
  const int half = lane >> 4, l16 = lane & 15;
  const float* __restrict__ arow = A + (size_t)(m0 + l16) * Cin;
  const float* __restrict__ wrow0 = W + (size_t)(n0 + l16) * Cin;
  const float* __restrict__ wrow1 = W + (size_t)(n0 + 16 + l16) * Cin;
  v8f acc0 = {0.f, 0.f, 0.f, 0.f, 0.f, 0.f, 0.f, 0.f};
  v8f acc1 = {0.f, 0.f, 0.f, 0.f, 0.f, 0.f, 0.f, 0.f};
  const int kofs = 2 * half;
  const int Cin4 = Cin & ~3;
#pragma unroll 2
  for (int k0 = 0; k0 < Cin4; k0 += 4) {
    const int k = k0 + kofs;
    v2f a, b0, b1;
    a.x  = arow[k];
    a.y  = arow[k + 1];
    b0.x = wrow0[k];
    b0.y = wrow0[k + 1];
    b1.x = wrow1[k];
    b1.y = wrow1[k + 1];
    acc0 = __builtin_amdgcn_wmma_f32_16x16x4_f32(false, a, false, b0,
                                                 (short)0, acc0, false, false);
    acc1 = __builtin_amdgcn_wmma_f32_16x16x4_f32(false, a, false, b1,
                                                 (short)0, acc1, false, false);
  }
  if (Cin4 < Cin) {  // single guarded tail step (Cin % 4 != 0)
    const int k = Cin4 + kofs;
    v2f a, b0, b1;
    a.x  = (k     < Cin) ? arow[k]      : 0.f;
    a.y  = (k + 1 < Cin) ? arow[k + 1]  : 0.f;
    b0.x = (k     < Cin) ? wrow0[k]     : 0.f;
    b0.y = (k + 1 < Cin) ? wrow0[k + 1] : 0.f;
    b1.x = (k     < Cin) ? wrow1[k]     : 0.f;
    b1.y = (k + 1 < Cin) ? wrow1[k + 1] : 0.f;
    acc0 = __builtin_amdgcn_wmma_f32_16x16x4_f32(false, a, false, b0,
                                                 (short)0, acc0, false, false);
    acc1 = __builtin_amdgcn_wmma_f32_16x16x4_f32(false, a, false, b1,
                                                 (short)0, acc1, false, false);
  }
  const float bv0 = bias[n0 + l16];
  const float bv1 = bias[n0 + 16 + l16];
#pragma unroll
  for (int r = 0; r < 8; ++r) {
    const int row = m0 + r + 8 * half;
    C[(size_t)row * Cout + (n0 + l16)]      = acc0[r] + bv0;
    C[(size_t)row * Cout + (n0 + 16 + l16)] = acc1[r] + bv1;
  }
}

// ---------------------------------------------------------------------------
// Farthest point sampling: one block per batch, dist cached in LDS.
// Matches jnp scan semantics: idx[0]=0, then argmax of running min-dist
// (first-index tie-break like jnp.argmax).
// ---------------------------------------------------------------------------
__global__ __launch_bounds__(256) void fps_kernel(
    const float* __restrict__ xyz, int* __restrict__ fidx, int N, int npoint) {
  __shared__ float dist[4096];
  __shared__ float rv[256];
  __shared__ int   ri[256];
  __shared__ float farpt[3];
  const int b = blockIdx.x;
  const float* __restrict__ px = xyz + (size_t)b * N * 3;
  const int t = threadIdx.x;
  for (int i = t; i < N; i += 256) dist[i] = 1e10f;
  __syncthreads();
  int far = 0;
  for (int it = 0; it < npoint; ++it) {
    if (t == 0) {
      fidx[b * npoint + it] = far;
      farpt[0] = px[far * 3 + 0];
      farpt[1] = px[far * 3 + 1];
      farpt[2] = px[far * 3 + 2];
    }
    __syncthreads();
    const float fx = farpt[0], fy = farpt[1], fz = farpt[2];
    float best = -1.0f;
    int bi = 0x7fffffff;
    for (int i = t; i < N; i += 256) {
      const float dx = px[i * 3 + 0] - fx;
      const float dy = px[i * 3 + 1] - fy;
      const float dz = px[i * 3 + 2] - fz;
      const float d = dx * dx + dy * dy + dz * dz;
      const float nd = fminf(dist[i], d);
      dist[i] = nd;
      if (nd > best) { best = nd; bi = i; }
    }
    rv[t] = best; ri[t] = bi;
    __syncthreads();
    for (int s = 128; s > 0; s >>= 1) {
      if (t < s) {
        if (rv[t + s] > rv[t] || (rv[t + s] == rv[t] && ri[t + s] < ri[t])) {
          rv[t] = rv[t + s]; ri[t] = ri[t + s];
        }
      }
      __syncthreads();
    }
    far = ri[0];
    __syncthreads();
  }
}

// gather new_xyz = xyz[fidx]
__global__ __launch_bounds__(256) void gather_xyz2_kernel(
    const float* __restrict__ xyz, const int* __restrict__ fidx,
    float* __restrict__ out, int N, int S, int B) {
  const int i = blockIdx.x * blockDim.x + threadIdx.x;
  if (i >= B * S) return;
  const int b = i / S;
  const int id = fidx[i];
  const float* __restrict__ p = xyz + ((size_t)b * N + id) * 3;
  float* __restrict__ o = out + (size_t)i * 3;
  o[0] = p[0]; o[1] = p[1]; o[2] = p[2];
}

// ---------------------------------------------------------------------------
// Ball query: one wave32 per query point; ordered compaction via ballot.
// Keeps first nsample in-radius indices in ascending order, pads with first.
// ---------------------------------------------------------------------------
__global__ __launch_bounds__(256) void ball_query_kernel(
    const float* __restrict__ xyz, const float* __restrict__ new_xyz,
    int* __restrict__ idx, int N, int S, int B, int nsample, float r2) {
  const int gw = (int)((blockIdx.x * blockDim.x + threadIdx.x) >> 5);
  const int lane = threadIdx.x & 31;
  if (gw >= B * S) return;
  const int b = gw / S;
  const float* __restrict__ q = new_xyz + (size_t)gw * 3;
  const float qx = q[0], qy = q[1], qz = q[2];
  const float* __restrict__ px = xyz + (size_t)b * N * 3;
  int* __restrict__ out = idx + (size_t)gw * nsample;
  int cnt = 0;
  int first = 0;
  bool have_first = false;
  for (int base = 0; base < N && cnt < nsample; base += 32) {
    const int i = base + lane;  // N is a multiple of 32
    const float dx = px[i * 3 + 0] - qx;
    const float dy = px[i * 3 + 1] - qy;
    const float dz = px[i * 3 + 2] - qz;
    const bool pred = (dx * dx + dy * dy + dz * dz) <= r2;
    const unsigned mask = (unsigned)(__ballot(pred) & 0xffffffffull);
    if (!have_first && mask) {
      first = base + __builtin_ctz(mask);
      have_first = true;
    }
    if (pred) {
      const int off = cnt + __builtin_popcount(mask & ((1u << lane) - 1u));
      if (off < nsample) out[off] = i;
    }
    cnt += __builtin_popcount(mask);
  }
  if (cnt > nsample) cnt = nsample;
  for (int j = cnt + lane; j < nsample; j += 32)
    out[j] = have_first ? first : 0;
}

// grouped = concat(xyz[idx] - new_xyz, points[idx]); Cp may be 0
__global__ __launch_bounds__(256) void group_kernel(
    const float* __restrict__ xyz, const float* __restrict__ new_xyz,
    const float* __restrict__ points, const int* __restrict__ idx,
    float* __restrict__ out, int B, int N, int S, int ns, int Cp) {
  const int C = 3 + Cp;
  const long long total = (long long)B * S * ns * C;
  long long e = (long long)blockIdx.x * blockDim.x + threadIdx.x;
  if (e >= total) return;
  const int c = (int)(e % C);
  const long long r = e / C;
  const int j = (int)(r % ns);
  const long long bs = r / ns;
  const int b = (int)(bs / S);
  const int id = idx[bs * ns + j];
  float v;
  if (c < 3) {
    v = xyz[((size_t)b * N + id) * 3 + c] - new_xyz[(size_t)bs * 3 + c];
  } else {
    v = points[((size_t)b * N + id) * Cp + (c - 3)];
  }
  out[e] = v;
}

// BN statistics: one block per channel, deterministic tree reduction.
__global__ __launch_bounds__(256) void bn_stats_kernel(
    const float* __restrict__ X, float* __restrict__ stats,
    long long rows, int C) {
  const int c = blockIdx.x;
  const int t = threadIdx.x;
  float s = 0.f, s2 = 0.f;
  for (long long r = t; r < rows; r += 256) {
    const float v = X[r * C + c];
    s += v; s2 += v * v;
  }
  __shared__ float sh[256], sh2[256];
  sh[t] = s; sh2[t] = s2;
  __syncthreads();
  for (int st = 128; st > 0; st >>= 1) {
    if (t < st) { sh[t] += sh[t + st]; sh2[t] += sh2[t + st]; }
    __syncthreads();
  }
  if (t == 0) { stats[c] = sh[0]; stats[C + c] = sh2[0]; }
}

__global__ __launch_bounds__(256) void bn_apply_kernel(
    const float* __restrict__ X, float* __restrict__ Y,
    const float* __restrict__ stats, const float* __restrict__ g,
    const float* __restrict__ be, long long rows, int C, int relu) {
  long long e = (long long)blockIdx.x * blockDim.x + threadIdx.x;
  if (e >= rows * C) return;
  const int c = (int)(e % C);
  const float mean = stats[c] / (float)rows;
  const float var = stats[C + c] / (float)rows - mean * mean;
  float v = (X[e] - mean) * rsqrtf(var + EPS_BN) * g[c] + be[c];
  if (relu) v = fmaxf(v, 0.f);
  Y[e] = v;
}

// final FC epilogue: BN (no relu) + transpose (B,N,C) -> (B,C,N)
__global__ __launch_bounds__(256) void bn_apply_transpose_kernel(
    const float* __restrict__ X, float* __restrict__ Y,
    const float* __restrict__ stats, const float* __restrict__ g,
    const float* __restrict__ be, int B, int N, int C) {
  const long long rows = (long long)B * N;
  long long e = (long long)blockIdx.x * blockDim.x + threadIdx.x;
  if (e >= rows * C) return;
  const int c = (int)(e % C);
  const long long r = e / C;
  const int n = (int)(r % N);
  const int b = (int)(r / N);
  const float mean = stats[c] / (float)rows;
  const float var = stats[C + c] / (float)rows - mean * mean;
  const float v = (X[e] - mean) * rsqrtf(var + EPS_BN) * g[c] + be[c];
  Y[((size_t)b * C + c) * N + n] = v;
}

__global__ __launch_bounds__(256) void maxpool_kernel(
    const float* __restrict__ X, float* __restrict__ Y, int BS, int ns, int C) {
  long long e = (long long)blockIdx.x * blockDim.x + threadIdx.x;
  if (e >= (long long)BS * C) return;
  const int c = (int)(e % C);
  const long long bs = e / C;
  const float* __restrict__ p = X + (size_t)bs * ns * C + c;
  float m = p[0];
  for (int j = 1; j < ns; ++j) m = fmaxf(m, p[(size_t)j * C]);
  Y[e] = m;
}

// sa3 input: concat(xyz(3), points(Cp)) per row (no centering)
__global__ __launch_bounds__(256) void concat_xyz_points_kernel(
    const float* __restrict__ xyz, const float* __restrict__ points,
    float* __restrict__ out, int rows, int Cp) {
  const int C = 3 + Cp;
  long long e = (long long)blockIdx.x * blockDim.x + threadIdx.x;
  if (e >= (long long)rows * C) return;
  const int c = (int)(e % C);
  const long long r = e / C;
  out[e] = (c < 3) ? xyz[r * 3 + c] : points[r * Cp + (c - 3)];
}

// fp3 input: concat(p2(C1 per row), broadcast p3(C2 per batch))
__global__ __launch_bounds__(256) void fp3_build_kernel(
    const float* __restrict__ p2, const float* __restrict__ p3,
    float* __restrict__ out, int B, int S, int C1, int C2) {
  const int C = C1 + C2;
  const long long total = (long long)B * S * C;
  long long e = (long long)blockIdx.x * blockDim.x + threadIdx.x;
  if (e >= total) return;
  const int c = (int)(e % C);
  const long long r = e / C;
  const int b = (int)(r / S);
  out[e] = (c < C1) ? p2[r * C1 + c] : p3[(size_t)b * C2 + (c - C1)];
}

// three nearest neighbors + inverse-distance weights
__global__ __launch_bounds__(256) void three_nn_kernel(
    const float* __restrict__ xyz1, const float* __restrict__ xyz2,
    int* __restrict__ idx3, float* __restrict__ w3, int B, int N1, int S2) {
  const int i = blockIdx.x * blockDim.x + threadIdx.x;
  if (i >= B * N1) return;
  const int b = i / N1;
  const float* __restrict__ q = xyz1 + (size_t)i * 3;
  const float qx = q[0], qy = q[1], qz = q[2];
  const float* __restrict__ p = xyz2 + (size_t)b * S2 * 3;
  float d0 = 1e30f, d1 = 1e30f, d2v = 1e30f;
  int i0 = 0, i1 = 0, i2 = 0;
  for (int s = 0; s < S2; ++s) {
    const float dx = p[s * 3 + 0] - qx;
    const float dy = p[s * 3 + 1] - qy;
    const float dz = p[s * 3 + 2] - qz;
    const float d = dx * dx + dy * dy + dz * dz;
    if (d < d0) { d2v = d1; i2 = i1; d1 = d0; i1 = i0; d0 = d; i0 = s; }
    else if (d < d1) { d2v = d1; i2 = i1; d1 = d; i1 = s; }
    else if (d < d2v) { d2v = d; i2 = s; }
  }
  const float r0 = 1.f / (d0 + 1e-8f);
  const float r1 = 1.f / (d1 + 1e-8f);
  const float r2 = 1.f / (d2v + 1e-8f);
  const float inv = 1.f / (r0 + r1 + r2);
  idx3[i * 3 + 0] = i0; idx3[i * 3 + 1] = i1; idx3[i * 3 + 2] = i2;
  w3[i * 3 + 0] = r0 * inv; w3[i * 3 + 1] = r1 * inv; w3[i * 3 + 2] = r2 * inv;
}

// fp input: concat(points1(C1), interp(points2 via 3-NN, C2))
__global__ __launch_bounds__(256) void fp_build_kernel(
    const float* __restrict__ points1, const float* __restrict__ points2,
    const int* __restrict__ idx3, const float* __restrict__ w3,
    float* __restrict__ out, int B, int N1, int S2, int C1, int C2) {
  const int C = C1 + C2;
  const long long total = (long long)B * N1 * C;
  long long e = (long long)blockIdx.x * blockDim.x + threadIdx.x;
  if (e >= total) return;
  const int c = (int)(e % C);
  const long long r = e / C;
  const int b = (int)(r / N1);
  float v;
  if (c < C1) {
    v = points1[r * C1 + c];
  } else {
    const int cc = c - C1;
    const float* __restrict__ pp = points2 + (size_t)b * S2 * C2;
    const int* __restrict__ id = idx3 + r * 3;
    const float* __restrict__ w = w3 + r * 3;
    v = w[0] * pp[(size_t)id[0] * C2 + cc] +
        w[1] * pp[(size_t)id[1] * C2 + cc] +
        w[2] * pp[(size_t)id[2] * C2 + cc];
  }
  out[e] = v;
}

// ---------------------------------------------------------------------------
// Host side
// ---------------------------------------------------------------------------
struct LayerP { const float *W, *b, *g, *be; int cout, cin; };

static void run_layer(const LayerP& L, const float* in, float* gemm_out,
                      float* final_out, long long rows, int relu,
                      float* stats, hipStream_t stream) {
  const long long tiles = (rows / 16) * (L.cout / 32);
  const int gblocks = (int)((tiles + 7) / 8);
  gemm_bias_wmma<<<gblocks, 256, 0, stream>>>(in, L.W, L.b, gemm_out,
                                              (int)rows, L.cin, L.cout);
  bn_stats_kernel<<<L.cout, 256, 0, stream>>>(gemm_out, stats, rows, L.cout);
  const long long tot = rows * L.cout;
  bn_apply_kernel<<<(int)((tot + 255) / 256), 256, 0, stream>>>(
      gemm_out, final_out, stats, L.g, L.be, rows, L.cout, relu);
}

extern "C" void kernel_launch(void* const* d_in, const int* in_sizes, int n_in,
                              void* d_out, int out_size, void* d_ws,
                              size_t ws_size, hipStream_t stream) {
  (void)in_sizes; (void)n_in; (void)out_size; (void)ws_size;
  const int B = 8, N = 4096;
  const float* xyz0 = (const float*)d_in[0];

  static const int co[17] = {64, 64, 128, 128, 128, 256, 256, 512, 1024,
                             256, 256, 256, 128, 128, 128, 128, 128};
  static const int ci[17] = {3, 64, 64, 131, 128, 128, 259, 256, 512,
                             1280, 256, 384, 256, 131, 128, 128, 128};
  LayerP L[17];
  for (int k = 0; k < 17; ++k) {
    L[k].W = (const float*)d_in[1 + 4 * k + 0];
    L[k].b = (const float*)d_in[1 + 4 * k + 1];
    L[k].g = (const float*)d_in[1 + 4 * k + 2];
    L[k].be = (const float*)d_in[1 + 4 * k + 3];
    L[k].cout = co[k]; L[k].cin = ci[k];
  }

  // workspace carve
  size_t off = 0;
  char* wsb = (char*)d_ws;
  auto carve = [&](size_t bytes) -> void* {
    void* p = wsb + off;
    off += (bytes + 255) & ~(size_t)255;
    return p;
  };
  float* bufA = (float*)carve((size_t)33554432 * 4);  // 262144 x 128 max
  float* bufB = (float*)carve((size_t)33554432 * 4);
  float* stats = (float*)carve((size_t)2048 * 4);
  int* fidx1 = (int*)carve((size_t)B * 512 * 4);
  float* nx1 = (float*)carve((size_t)B * 512 * 3 * 4);
  int* idx1 = (int*)carve((size_t)B * 512 * 64 * 4);
  int* fidx2 = (int*)carve((size_t)B * 128 * 4);
  float* nx2 = (float*)carve((size_t)B * 128 * 3 * 4);
  int* idx2 = (int*)carve((size_t)B * 128 * 64 * 4);
  float* p1 = (float*)carve((size_t)B * 512 * 128 * 4);
  float* p2 = (float*)carve((size_t)B * 128 * 256 * 4);
  float* p3 = (float*)carve((size_t)B * 1024 * 4);
  float* q2 = (float*)carve((size_t)B * 128 * 256 * 4);
  float* q1 = (float*)carve((size_t)B * 512 * 128 * 4);
  float* q0 = (float*)carve((size_t)B * 4096 * 128 * 4);
  int* nn_idx = (int*)carve((size_t)B * 4096 * 3 * 4);
  float* nn_w = (float*)carve((size_t)B * 4096 * 3 * 4);

  // ----- SA1: N=4096 -> 512 centers, 64 samples, r=0.2, MLP 3->64->64->128
  fps_kernel<<<B, 256, 0, stream>>>(xyz0, fidx1, N, 512);
  gather_xyz2_kernel<<<(B * 512 + 255) / 256, 256, 0, stream>>>(xyz0, fidx1,
                                                                nx1, N, 512, B);
  ball_query_kernel<<<(B * 512 * 32 + 255) / 256, 256, 0, stream>>>(
      xyz0, nx1, idx1, N, 512, B, 64, 0.04f);
  {
    const long long tot = (long long)B * 512 * 64 * 3;
    group_kernel<<<(int)((tot + 255) / 256), 256, 0, stream>>>(
        xyz0, nx1, (const float*)nullptr, idx1, bufA, B, N, 512, 64, 0);
  }
  const long long rows1 = (long long)B * 512 * 64;
  run_layer(L[0], bufA, bufB, bufB, rows1, 1, stats, stream);
  run_layer(L[1], bufB, bufA, bufA, rows1, 1, stats, stream);
  run_layer(L[2], bufA, bufB, bufB, rows1, 1, stats, stream);
  {
    const long long tot = (long long)B * 512 * 128;
    maxpool_kernel<<<(int)((tot + 255) / 256), 256, 0, stream>>>(bufB, p1,
                                                                 B * 512, 64, 128);
  }

  // ----- SA2: 512 -> 128 centers, 64 samples, r=0.4, MLP 131->128->128->256
  fps_kernel<<<B, 256, 0, stream>>>(nx1, fidx2, 512, 128);
  gather_xyz2_kernel<<<(B * 128 + 255) / 256, 256, 0, stream>>>(nx1, fidx2,
                                                                nx2, 512, 128, B);
  ball_query_kernel<<<(B * 128 * 32 + 255) / 256, 256, 0, stream>>>(
      nx1, nx2, idx2, 512, 128, B, 64, 0.16f);
  {
    const long long tot = (long long)B * 128 * 64 * 131;
    group_kernel<<<(int)((tot + 255) / 256), 256, 0, stream>>>(
        nx1, nx2, p1, idx2, bufA, B, 512, 128, 64, 128);
  }
  const long long rows2 = (long long)B * 128 * 64;
  run_layer(L[3], bufA, bufB, bufB, rows2, 1, stats, stream);
  run_layer(L[4], bufB, bufA, bufA, rows2, 1, stats, stream);
  run_layer(L[5], bufA, bufB, bufB, rows2, 1, stats, stream);
  {
    const long long tot = (long long)B * 128 * 256;
    maxpool_kernel<<<(int)((tot + 255) / 256), 256, 0, stream>>>(bufB, p2,
                                                                 B * 128, 64, 256);
  }

  // ----- SA3 (group-all): rows = B*128, C=259, MLP 259->256->512->1024
  {
    const long long tot = (long long)B * 128 * 259;
    concat_xyz_points_kernel<<<(int)((tot + 255) / 256), 256, 0, stream>>>(
        nx2, p2, bufA, B * 128, 256);
  }
  const long long rows3 = (long long)B * 128;
  run_layer(L[6], bufA, bufB, bufB, rows3, 1, stats, stream);
  run_layer(L[7], bufB, bufA, bufA, rows3, 1, stats, stream);
  run_layer(L[8], bufA, bufB, bufB, rows3, 1, stats, stream);
  {
    const long long tot = (long long)B * 1024;
    maxpool_kernel<<<(int)((tot + 255) / 256), 256, 0, stream>>>(bufB, p3, B,
                                                                 128, 1024);
  }

  // ----- FP3: broadcast p3 + p2 -> MLP 1280->256->256 => q2
  {
    const long long tot = (long long)B * 128 * 1280;
    fp3_build_kernel<<<(int)((tot + 255) / 256), 256, 0, stream>>>(
        p2, p3, bufA, B, 128, 256, 1024);
  }
  run_layer(L[9], bufA, bufB, bufB, rows3, 1, stats, stream);
  run_layer(L[10], bufB, bufA, q2, rows3, 1, stats, stream);

  // ----- FP2: interp(q2 @ 128 pts) to 512 pts + p1 -> MLP 384->256->128 => q1
  three_nn_kernel<<<(B * 512 + 255) / 256, 256, 0, stream>>>(nx1, nx2, nn_idx,
                                                             nn_w, B, 512, 128);
  {
    const long long tot = (long long)B * 512 * 384;
    fp_build_kernel<<<(int)((tot + 255) / 256), 256, 0, stream>>>(
        p1, q2, nn_idx, nn_w, bufA, B, 512, 128, 128, 256);
  }
  const long long rows4 = (long long)B * 512;
  run_layer(L[11], bufA, bufB, bufB, rows4, 1, stats, stream);
  run_layer(L[12], bufB, bufA, q1, rows4, 1, stats, stream);

  // ----- FP1: interp(q1 @ 512 pts) to 4096 pts + xyz0 -> 131->128->128->128 => q0
  three_nn_kernel<<<(B * 4096 + 255) / 256, 256, 0, stream>>>(
      xyz0, nx1, nn_idx, nn_w, B, 4096, 512);
  {
    const long long tot = (long long)B * 4096 * 131;
    fp_build_kernel<<<(int)((tot + 255) / 256), 256, 0, stream>>>(
        xyz0, q1, nn_idx, nn_w, bufA, B, 4096, 512, 3, 128);
  }
  const long long rows5 = (long long)B * 4096;
  run_layer(L[13], bufA, bufB, bufB, rows5, 1, stats, stream);
  run_layer(L[14], bufB, bufA, bufA, rows5, 1, stats, stream);
  run_layer(L[15], bufA, bufB, q0, rows5, 1, stats, stream);

  // ----- FC: q0 * W^T + b, BN (no relu), transpose to (B,128,N) into d_out
  {
    const long long tiles = (rows5 / 16) * (128 / 32);
    gemm_bias_wmma<<<(int)((tiles + 7) / 8), 256, 0, stream>>>(
        q0, L[16].W, L[16].b, bufA, (int)rows5, 128, 128);
    bn_stats_kernel<<<128, 256, 0, stream>>>(bufA, stats, rows5, 128);
    const long long tot = rows5 * 128;
    bn_apply_transpose_kernel<<<(int)((tot + 255) / 256), 256, 0, stream>>>(
        bufA, (float*)d_out, stats, L[16].g, L[16].be, B, N, 128);
  }
  // append p3 (transpose of (B,1,1024) is a flat no-op)
  hipMemcpyAsync((float*)d_out + (size_t)B * 128 * N, p3,
                 (size_t)B * 1024 * sizeof(float), hipMemcpyDeviceToDevice,
                 stream);
}